// EEGGATConvNet_8993661518687
// MI455X (gfx1250) — compile-verified
//
#include <hip/hip_runtime.h>
#include <math.h>

#define N_NODES 131072
#define N_EDGES 2097152
#define N_GRAPHS 16384
#define LRELU_ATT 0.2f
#define LRELU_ACT 0.01f
#define BN_EPS 1e-5f

typedef float v2f __attribute__((ext_vector_type(2)));
typedef float v8f __attribute__((ext_vector_type(8)));

__device__ __forceinline__ float lrelu(float x, float s) { return x > 0.0f ? x : s * x; }

// float atomic max via signed-max / unsigned-min bit trick (monotone over IEEE order)
__device__ __forceinline__ void atomicMaxF(float* addr, float val) {
    int vi = __float_as_int(val);
    if (vi >= 0) atomicMax((int*)addr, vi);
    else         atomicMin((unsigned int*)addr, (unsigned int)vi);
}

// pad x (N x 6) -> xpad (N x 8) with zero cols 6,7 so GEMM A-loads are unguarded b64
__global__ void pad_x_kernel(const float* __restrict__ x, float* __restrict__ xpad)
{
    int gid = blockIdx.x * blockDim.x + threadIdx.x;
    if (gid >= N_NODES * 8) return;
    int n = gid >> 3, c = gid & 7;
    xpad[gid] = (c < 6) ? x[n * 6 + c] : 0.0f;
}

// ---------------------------------------------------------------------------
// Dense GEMM Y[N x ldy] = X[N x Kpad] @ W[Ktrue x P]   (padded cols/rows -> 0)
// W staged in LDS pair-packed: (k,col) -> wlds[(k>>1)*2*ldy + 2*col + (k&1)]
// so both WMMA operands are single unguarded 64-bit loads.
// One wave32 per 16x16 tile; fp32 WMMA 16x16x4 accumulate.
// A layout (16x4 f32): lane l -> M = l&15, K = 2*(l>>4) + vgpr
// B layout (4x16 f32): lane l -> N = l&15, K = 2*(l>>4) + vgpr
// C/D layout: vgpr i, lane l -> M = i + 8*(l>>4), N = l&15
// ---------------------------------------------------------------------------
__global__ __launch_bounds__(256) void gemm_wmma_f32(
    const float* __restrict__ X, int ldx, int Ktrue, int Kpad,
    const float* __restrict__ W, int P,
    float* __restrict__ Y, int ldy, int nrows)
{
    extern __shared__ float wlds[];   // Kpad * ldy floats
    const int tid = threadIdx.x;
    const int tilesP = ldy >> 4;

    // stage + zero-pad W into LDS (guards only here, outside the hot loop)
    const int total = Kpad * ldy;
    for (int i = tid; i < total; i += blockDim.x) {
        int k = i / ldy, col = i - k * ldy;
        float v = (k < Ktrue && col < P) ? W[k * P + col] : 0.0f;
        wlds[(k >> 1) * (ldy << 1) + (col << 1) + (k & 1)] = v;
    }
    __syncthreads();

    const int lane  = tid & 31;
    const int wave  = blockIdx.x * (blockDim.x >> 5) + (tid >> 5);
    const int tilesM = nrows >> 4;
    if (wave >= tilesM * tilesP) return;   // wave-uniform: EXEC all-ones at WMMA
    const int tp = wave % tilesP;
    const int tm = wave / tilesP;
    const int r    = lane & 15;
    const int col  = tp * 16 + r;
    const int khalf = (lane >> 4) << 1;
    const float* xrow = X + (size_t)(tm * 16 + r) * ldx + khalf;
    const float* wptr = wlds + (khalf >> 1) * (ldy << 1) + (col << 1);
    const int wstep = ldy << 2;           // advance 4 k's = 2 pair-rows

    v8f acc = {};
    for (int k0 = 0; k0 < Kpad; k0 += 4) {
        v2f a = *(const v2f*)(xrow + k0);
        v2f b = *(const v2f*)(wptr);
        wptr += wstep;
        acc = __builtin_amdgcn_wmma_f32_16x16x4_f32(false, a, false, b, (short)0, acc,
                                                    false, false);
    }
    const int mbase = tm * 16 + ((lane >> 4) << 3);
#pragma unroll
    for (int i = 0; i < 8; ++i)
        Y[(size_t)(mbase + i) * ldy + col] = acc[i];
}

// alpha_src[n,h] = sum_c h[n,h,c]*a_src[h,c] ; same for alpha_dst (float4 main loop)
__global__ void alpha_node_kernel(const float* __restrict__ h, int ldh, int H, int C,
                                  const float* __restrict__ asrc,
                                  const float* __restrict__ adst,
                                  float* __restrict__ out_s, float* __restrict__ out_d)
{
    int gid = blockIdx.x * blockDim.x + threadIdx.x;
    if (gid >= N_NODES * H) return;
    int node = gid / H, hh = gid - node * H;
    const float* hp = h + (size_t)node * ldh + hh * C;
    const float* ap = asrc + hh * C;
    const float* bp = adst + hh * C;
    float s = 0.0f, d = 0.0f;
    int c = 0;
    for (; c + 4 <= C; c += 4) {
        float4 v = *(const float4*)(hp + c);
        s += v.x * ap[c] + v.y * ap[c + 1] + v.z * ap[c + 2] + v.w * ap[c + 3];
        d += v.x * bp[c] + v.y * bp[c + 1] + v.z * bp[c + 2] + v.w * bp[c + 3];
    }
    for (; c < C; ++c) { float v = hp[c]; s += v * ap[c]; d += v * bp[c]; }
    out_s[gid] = s;
    out_d[gid] = d;
}

// aeconst[h] = sum_c We[0,h*C+c] * ae[h,c]   (edge feature is scalar ew[e])
__global__ void ae_const_kernel(const float* __restrict__ We, const float* __restrict__ ae,
                                float* __restrict__ aec, int H, int C)
{
    int h = threadIdx.x;
    if (h >= H) return;
    float s = 0.0f;
    for (int c = 0; c < C; ++c) s += We[h * C + c] * ae[h * C + c];
    aec[h] = s;
}

__global__ void fill_kernel(float* __restrict__ p, float v, int n)
{
    int gid = blockIdx.x * blockDim.x + threadIdx.x;
    if (gid < n) p[gid] = v;
}

// pass A: alpha = lrelu(asrc[src]+adst[dst]+ew*aec, 0.2); store; segment max over dst
__global__ void edge_alpha_max_kernel(const int* __restrict__ ei, const float* __restrict__ ew,
                                      const float* __restrict__ asrc,
                                      const float* __restrict__ adst,
                                      const float* __restrict__ aec,
                                      float* __restrict__ albuf, float* __restrict__ amax, int H)
{
    int gid = blockIdx.x * blockDim.x + threadIdx.x;
    if (gid >= N_EDGES * H) return;
    int e = gid / H, h = gid - e * H;
    int s = ei[e], d = ei[N_EDGES + e];
    float a = asrc[s * H + h] + adst[d * H + h] + ew[e] * aec[h];
    a = lrelu(a, LRELU_ATT);
    albuf[gid] = a;
    atomicMaxF(&amax[d * H + h], a);
}

// pass B: ex = exp(alpha - amax[dst]); store; segment sum over dst
__global__ void edge_exp_sum_kernel(const int* __restrict__ ei, float* __restrict__ albuf,
                                    const float* __restrict__ amax, float* __restrict__ denom,
                                    int H)
{
    int gid = blockIdx.x * blockDim.x + threadIdx.x;
    if (gid >= N_EDGES * H) return;
    int e = gid / H, h = gid - e * H;
    int d = ei[N_EDGES + e];
    float ex = expf(albuf[gid] - amax[d * H + h]);
    albuf[gid] = ex;
    atomicAdd(&denom[d * H + h], ex);
}

// pass C: out[dst] += h[src] * (ex / (denom[dst]+1e-16)), 4 channels per thread
__global__ void edge_aggregate_kernel(const int* __restrict__ ei,
                                      const float* __restrict__ albuf,
                                      const float* __restrict__ denom,
                                      const float* __restrict__ htmp, int ldh,
                                      float* __restrict__ out, int H, int C, int HC, int TPE)
{
    long long gid = (long long)blockIdx.x * blockDim.x + threadIdx.x;
    int e = (int)(gid / TPE);
    if (e >= N_EDGES) return;
    int t  = (int)(gid - (long long)e * TPE);
    int c0 = t * 4;
    int s = ei[e], d = ei[N_EDGES + e];
    int head = c0 / C;
    float att = albuf[(size_t)e * H + head] / (denom[d * H + head] + 1e-16f);
    const float4 hv = *(const float4*)(htmp + (size_t)s * ldh + c0);
    float* op = out + (size_t)d * ldh + c0;
    if (c0 + 4 <= HC) {                         // fast path (always for HC=64/128)
        atomicAdd(op + 0, hv.x * att);
        atomicAdd(op + 1, hv.y * att);
        atomicAdd(op + 2, hv.z * att);
        atomicAdd(op + 3, hv.w * att);
    } else {
        if (c0 + 0 < HC) atomicAdd(op + 0, hv.x * att);
        if (c0 + 1 < HC) atomicAdd(op + 1, hv.y * att);
        if (c0 + 2 < HC) atomicAdd(op + 2, hv.z * att);
        if (c0 + 3 < HC) atomicAdd(op + 3, hv.w * att);
    }
}

// layers 1-3 epilogue: out = lrelu(out + b, 0.01), float4 (ldh == HC, HC % 4 == 0)
__global__ void bias_act_kernel(float* __restrict__ out, const float* __restrict__ b, int HC)
{
    int gid = blockIdx.x * blockDim.x + threadIdx.x;
    if (gid >= (N_NODES * HC) >> 2) return;
    int base = gid << 2;
    int c = base % HC;
    float4 v = *(const float4*)(out + base);
    v.x = lrelu(v.x + b[c + 0], LRELU_ACT);
    v.y = lrelu(v.y + b[c + 1], LRELU_ACT);
    v.z = lrelu(v.z + b[c + 2], LRELU_ACT);
    v.w = lrelu(v.w + b[c + 3], LRELU_ACT);
    *(float4*)(out + base) = v;
}

// layer-4 epilogue: +b4, batchnorm, lrelu, segment-sum into per-graph buffer
__global__ void bn_pool_kernel(const float* __restrict__ out4, const float* __restrict__ b4,
                               const float* __restrict__ bg, const float* __restrict__ bb,
                               const float* __restrict__ rm, const float* __restrict__ rv,
                               const int* __restrict__ batch, float* __restrict__ gbuf)
{
    int gid = blockIdx.x * blockDim.x + threadIdx.x;
    if (gid >= N_NODES * 50) return;
    int n = gid / 50, c = gid - n * 50;
    float v = out4[(size_t)n * 64 + c] + b4[c];
    v = (v - rm[c]) * rsqrtf(rv[c] + BN_EPS) * bg[c] + bb[c];
    v = lrelu(v, LRELU_ACT);
    atomicAdd(&gbuf[batch[n] * 50 + c], v);
}

// final MLP 50 -> 30 -> 20 -> 2, one thread per graph
__global__ void mlp_kernel(const float* __restrict__ gbuf,
                           const float* __restrict__ Wf1, const float* __restrict__ bf1,
                           const float* __restrict__ Wf2, const float* __restrict__ bf2,
                           const float* __restrict__ Wf3, const float* __restrict__ bf3,
                           float* __restrict__ out)
{
    int gid = blockIdx.x * blockDim.x + threadIdx.x;
    if (gid >= N_GRAPHS) return;
    const float* gr = gbuf + (size_t)gid * 50;
    float t1[30];
    for (int j = 0; j < 30; ++j) {
        float s = bf1[j];
        for (int i = 0; i < 50; ++i) s += gr[i] * Wf1[i * 30 + j];
        t1[j] = lrelu(s, LRELU_ACT);
    }
    float t2[20];
    for (int j = 0; j < 20; ++j) {
        float s = bf2[j];
        for (int i = 0; i < 30; ++i) s += t1[i] * Wf2[i * 20 + j];
        t2[j] = lrelu(s, LRELU_ACT);
    }
    for (int j = 0; j < 2; ++j) {
        float s = bf3[j];
        for (int i = 0; i < 20; ++i) s += t2[i] * Wf3[i * 2 + j];
        out[gid * 2 + j] = s;
    }
}

// ---------------------------------------------------------------------------
static inline unsigned cdiv(long long a, long long b) { return (unsigned)((a + b - 1) / b); }

static void run_gat_layer(hipStream_t stream,
                          const float* X, int ldx, int Ktrue, int Kpad,
                          const float* W, const float* as_, const float* ad_,
                          const float* We_, const float* ae_, const float* b_,
                          int H, int C, int ldh,
                          const int* ei, const float* ew,
                          float* htmp, float* out,
                          float* asrcb, float* adstb, float* amax, float* denom,
                          float* aec, float* albuf, bool do_act)
{
    const int HC  = H * C;
    const int TPE = (HC + 3) / 4;
    const long long waves = (long long)(N_NODES / 16) * (ldh / 16);
    const size_t ldsBytes = (size_t)Kpad * ldh * sizeof(float);

    gemm_wmma_f32<<<cdiv(waves, 8), 256, ldsBytes, stream>>>(
        X, ldx, Ktrue, Kpad, W, HC, htmp, ldh, N_NODES);
    alpha_node_kernel<<<cdiv((long long)N_NODES * H, 256), 256, 0, stream>>>(
        htmp, ldh, H, C, as_, ad_, asrcb, adstb);
    ae_const_kernel<<<1, 32, 0, stream>>>(We_, ae_, aec, H, C);
    fill_kernel<<<cdiv((long long)N_NODES * H, 256), 256, 0, stream>>>(amax, -INFINITY, N_NODES * H);
    fill_kernel<<<cdiv((long long)N_NODES * H, 256), 256, 0, stream>>>(denom, 0.0f, N_NODES * H);
    fill_kernel<<<cdiv((long long)N_NODES * ldh, 256), 256, 0, stream>>>(out, 0.0f, N_NODES * ldh);
    edge_alpha_max_kernel<<<cdiv((long long)N_EDGES * H, 256), 256, 0, stream>>>(
        ei, ew, asrcb, adstb, aec, albuf, amax, H);
    edge_exp_sum_kernel<<<cdiv((long long)N_EDGES * H, 256), 256, 0, stream>>>(
        ei, albuf, amax, denom, H);
    edge_aggregate_kernel<<<cdiv((long long)N_EDGES * TPE, 256), 256, 0, stream>>>(
        ei, albuf, denom, htmp, ldh, out, H, C, HC, TPE);
    if (do_act)
        bias_act_kernel<<<cdiv((long long)N_NODES * HC / 4, 256), 256, 0, stream>>>(out, b_, HC);
}

extern "C" void kernel_launch(void* const* d_in, const int* in_sizes, int n_in,
                              void* d_out, int out_size, void* d_ws, size_t ws_size,
                              hipStream_t stream)
{
    const float* x     = (const float*)d_in[0];
    const int*   ei    = (const int*)d_in[1];
    const float* ew    = (const float*)d_in[2];
    const int*   batch = (const int*)d_in[3];
    const float* W1  = (const float*)d_in[4];
    const float* as1 = (const float*)d_in[5];
    const float* ad1 = (const float*)d_in[6];
    const float* We1 = (const float*)d_in[7];
    const float* ae1 = (const float*)d_in[8];
    const float* b1  = (const float*)d_in[9];
    const float* W2  = (const float*)d_in[10];
    const float* as2 = (const float*)d_in[11];
    const float* ad2 = (const float*)d_in[12];
    const float* We2 = (const float*)d_in[13];
    const float* ae2 = (const float*)d_in[14];
    const float* b2  = (const float*)d_in[15];
    const float* W3  = (const float*)d_in[16];
    const float* as3 = (const float*)d_in[17];
    const float* ad3 = (const float*)d_in[18];
    const float* We3 = (const float*)d_in[19];
    const float* ae3 = (const float*)d_in[20];
    const float* b3  = (const float*)d_in[21];
    const float* W4  = (const float*)d_in[22];
    const float* as4 = (const float*)d_in[23];
    const float* ad4 = (const float*)d_in[24];
    const float* We4 = (const float*)d_in[25];
    const float* ae4 = (const float*)d_in[26];
    const float* b4  = (const float*)d_in[27];
    const float* bn_g  = (const float*)d_in[28];
    const float* bn_b  = (const float*)d_in[29];
    const float* bn_rm = (const float*)d_in[30];
    const float* bn_rv = (const float*)d_in[31];
    const float* Wf1 = (const float*)d_in[32];
    const float* bf1 = (const float*)d_in[33];
    const float* Wf2 = (const float*)d_in[34];
    const float* bf2 = (const float*)d_in[35];
    const float* Wf3 = (const float*)d_in[36];
    const float* bf3 = (const float*)d_in[37];
    (void)in_sizes; (void)n_in; (void)out_size; (void)ws_size;

    // workspace layout (bytes)
    char* ws = (char*)d_ws;
    float* hA    = (float*)(ws + 0);                    // 131072*128 floats (64 MB)
    float* hB    = (float*)(ws + (size_t)67108864);     // 131072*128 floats (64 MB)
    float* albuf = (float*)(ws + (size_t)134217728);    // 2097152*4 floats (32 MB)
    float* asrcb = (float*)(ws + (size_t)167772160);    // 131072*4
    float* adstb = (float*)(ws + (size_t)169869312);    // 131072*4
    float* amax  = (float*)(ws + (size_t)171966464);    // 131072*4
    float* denom = (float*)(ws + (size_t)174063616);    // 131072*4
    float* aec   = (float*)(ws + (size_t)176160768);    // 64 floats
    float* gbuf  = (float*)(ws + (size_t)176161024);    // 16384*50
    float* xpad  = (float*)(ws + (size_t)179437824);    // 131072*8 (4 MB)

    pad_x_kernel<<<cdiv((long long)N_NODES * 8, 256), 256, 0, stream>>>(x, xpad);

    // GAT layer 1: in 6 (padded 8) -> H4 x C16 (concat 64), out -> hB
    run_gat_layer(stream, xpad, 8, 6, 8, W1, as1, ad1, We1, ae1, b1, 4, 16, 64,
                  ei, ew, hA, hB, asrcb, adstb, amax, denom, aec, albuf, true);
    // GAT layer 2: 64 -> H4 x C32 (concat 128)
    run_gat_layer(stream, hB, 64, 64, 64, W2, as2, ad2, We2, ae2, b2, 4, 32, 128,
                  ei, ew, hA, hB, asrcb, adstb, amax, denom, aec, albuf, true);
    // GAT layer 3: 128 -> H4 x C16 (concat 64)
    run_gat_layer(stream, hB, 128, 128, 128, W3, as3, ad3, We3, ae3, b3, 4, 16, 64,
                  ei, ew, hA, hB, asrcb, adstb, amax, denom, aec, albuf, true);
    // GAT layer 4: 64 -> H1 x C50 (mean over 1 head == identity); padded stride 64
    run_gat_layer(stream, hB, 64, 64, 64, W4, as4, ad4, We4, ae4, b4, 1, 50, 64,
                  ei, ew, hA, hB, asrcb, adstb, amax, denom, aec, albuf, false);

    // bias + batchnorm + lrelu + graph pooling
    fill_kernel<<<cdiv((long long)N_GRAPHS * 50, 256), 256, 0, stream>>>(gbuf, 0.0f, N_GRAPHS * 50);
    bn_pool_kernel<<<cdiv((long long)N_NODES * 50, 256), 256, 0, stream>>>(
        hB, b4, bn_g, bn_b, bn_rm, bn_rv, batch, gbuf);
    // final MLP
    mlp_kernel<<<cdiv(N_GRAPHS, 256), 256, 0, stream>>>(
        gbuf, Wf1, bf1, Wf2, bf2, Wf3, bf3, (float*)d_out);
}